// GFNCodebook_27315992003198
// MI455X (gfx1250) — compile-verified
//
#include <hip/hip_runtime.h>

// Problem constants (match reference)
#define B_ 4096
#define S_ 32
#define K_ 1024
#define D_ 128
#define DECAY_ 0.99f
#define OMD_   0.01f     // 1 - DECAY
#define EPS_   1e-5f

typedef __attribute__((ext_vector_type(2))) float v2f;
typedef __attribute__((ext_vector_type(8))) float v8f;

// ---------------------------------------------------------------------------
// K1: per-s histogram of indices via LDS float atomics (ds_add_f32).
// One block per s handles all B -> no global atomics, deterministic.
// ---------------------------------------------------------------------------
__global__ void vq_hist(const int* __restrict__ indices, float* __restrict__ counts) {
    __shared__ float bins[K_];
    const int s = blockIdx.x;
    for (int k = threadIdx.x; k < K_; k += blockDim.x) bins[k] = 0.0f;
    __syncthreads();
    for (int b = threadIdx.x; b < B_; b += blockDim.x) {
        const int k = indices[b * S_ + s];
        atomicAdd(&bins[k], 1.0f);
    }
    __syncthreads();
    for (int k = threadIdx.x; k < K_; k += blockDim.x)
        counts[(size_t)s * K_ + k] = bins[k];
}

// ---------------------------------------------------------------------------
// K2: z_q gather. One wave32 per (b,s) row; each lane moves one float4 (B128).
// Row index r = b*S + s, so s = r % 32. Embedding row is 512B, L2-resident.
// ---------------------------------------------------------------------------
__global__ void vq_gather(const int* __restrict__ indices,
                          const float* __restrict__ emb,
                          float* __restrict__ zq) {
    const int row  = blockIdx.x * 8 + (threadIdx.x >> 5);
    const int lane = threadIdx.x & 31;
    const int s = row & (S_ - 1);
    const int k = indices[row];
    const float4* __restrict__ src =
        (const float4*)(emb + ((size_t)s * K_ + (size_t)k) * D_);
    float4* __restrict__ dst = (float4*)(zq + (size_t)row * D_);
    dst[lane] = src[lane];
}

// ---------------------------------------------------------------------------
// K3: per-s new_ecs + row-sum n[s] via V_WMMA_F32_16X16X4_F32 ones-matmul,
// then inverse Laplace-smoothed cluster size.
// One wave32 per s (EXEC all ones, as WMMA requires).
// A (16x4) carries 64 fresh values per WMMA (2 per lane); B == 1 makes
// D[m,n] = sum_k A[m,k], so sum over one column of the accumulator equals
// the sum of all 1024 inputs regardless of (M,K) placement.
// ---------------------------------------------------------------------------
__global__ void vq_ecs(const float* __restrict__ ecs,
                       const float* __restrict__ counts,
                       float* __restrict__ necs,
                       float* __restrict__ invcs) {
    const int s    = blockIdx.x;
    const int lane = threadIdx.x;            // blockDim.x == 32
    const float* __restrict__ e = ecs    + (size_t)s * K_;
    const float* __restrict__ c = counts + (size_t)s * K_;
    float* __restrict__ no = necs  + (size_t)s * K_;
    float* __restrict__ io = invcs + (size_t)s * K_;

    v2f vals[16];
    v8f acc = {};
    v2f ones; ones.x = 1.0f; ones.y = 1.0f;

    #pragma unroll
    for (int t = 0; t < 16; ++t) {
        const int base = t * 64 + lane * 2;
        const float2 ev = *(const float2*)(e + base);
        const float2 cv = *(const float2*)(c + base);
        v2f nv;
        nv.x = DECAY_ * ev.x + OMD_ * cv.x;
        nv.y = DECAY_ * ev.y + OMD_ * cv.y;
        vals[t] = nv;
        *(float2*)(no + base) = make_float2(nv.x, nv.y);
        // (neg_a, A, neg_b, B, c_mod, C, reuse_a, reuse_b)
        acc = __builtin_amdgcn_wmma_f32_16x16x4_f32(
            false, nv, false, ones, (short)0, acc, false, false);
    }

    // Column sum of accumulator: lane L holds D[0..7, L%16] (L<16) or
    // D[8..15, L-16] (L>=16); pair-sum across the half-wave gives n.
    float partial = acc[0] + acc[1] + acc[2] + acc[3]
                  + acc[4] + acc[5] + acc[6] + acc[7];
    const float n   = partial + __shfl_xor(partial, 16, 32);
    const float num = n + (float)K_ * EPS_;   // n + K*eps

    #pragma unroll
    for (int t = 0; t < 16; ++t) {
        const int base = t * 64 + lane * 2;
        float2 iv;
        iv.x = num / ((vals[t].x + EPS_) * n);
        iv.y = num / ((vals[t].y + EPS_) * n);
        *(float2*)(io + base) = iv;
    }
}

// ---------------------------------------------------------------------------
// K4: streaming EMA update + normalized embedding.
// One wave32 per (s,k) row; float4 per lane. dw collapses to counts*embedding.
// ---------------------------------------------------------------------------
__global__ void vq_ema(const float* __restrict__ emb,
                       const float* __restrict__ emaw,
                       const float* __restrict__ counts,
                       const float* __restrict__ invcs,
                       float* __restrict__ neww,
                       float* __restrict__ newemb) {
    const int row  = blockIdx.x * 8 + (threadIdx.x >> 5);
    const int lane = threadIdx.x & 31;
    const float cs  = OMD_ * counts[row];
    const float inv = invcs[row];
    const float4 e = ((const float4*)(emb  + (size_t)row * D_))[lane];
    const float4 w = ((const float4*)(emaw + (size_t)row * D_))[lane];
    float4 nw, ne;
    nw.x = DECAY_ * w.x + cs * e.x;  ne.x = nw.x * inv;
    nw.y = DECAY_ * w.y + cs * e.y;  ne.y = nw.y * inv;
    nw.z = DECAY_ * w.z + cs * e.z;  ne.z = nw.z * inv;
    nw.w = DECAY_ * w.w + cs * e.w;  ne.w = nw.w * inv;
    ((float4*)(neww   + (size_t)row * D_))[lane] = nw;
    ((float4*)(newemb + (size_t)row * D_))[lane] = ne;
}

// ---------------------------------------------------------------------------
extern "C" void kernel_launch(void* const* d_in, const int* in_sizes, int n_in,
                              void* d_out, int out_size, void* d_ws, size_t ws_size,
                              hipStream_t stream) {
    const int*   indices = (const int*)  d_in[0];   // [B,S]
    const float* emb     = (const float*)d_in[1];   // [S,K,D]
    const float* ecs     = (const float*)d_in[2];   // [S,K]
    const float* emaw    = (const float*)d_in[3];   // [S,K,D]

    float* out    = (float*)d_out;
    float* zq     = out;                                       // [B,S,D]
    float* newemb = zq     + (size_t)B_ * S_ * D_;             // [S,K,D]
    float* necs   = newemb + (size_t)S_ * K_ * D_;             // [S,K]
    float* neww   = necs   + (size_t)S_ * K_;                  // [S,K,D]

    float* counts = (float*)d_ws;                              // S*K floats
    float* invcs  = counts + (size_t)S_ * K_;                  // S*K floats

    // counts (fully rewritten each call; no stale-ws dependence)
    vq_hist<<<S_, 256, 0, stream>>>(indices, counts);
    // z_q gather (independent of histogram)
    vq_gather<<<(B_ * S_) / 8, 256, 0, stream>>>(indices, emb, zq);
    // new_ecs + n[s] (WMMA reduction) + inverse cluster size
    vq_ecs<<<S_, 32, 0, stream>>>(ecs, counts, necs, invcs);
    // new_ema_w + new_embedding
    vq_ema<<<(S_ * K_) / 8, 256, 0, stream>>>(emb, emaw, counts, invcs, neww, newemb);
}